// CSAM_38534446580009
// MI455X (gfx1250) — compile-verified
//
#include <hip/hip_runtime.h>
#include <math.h>

// ---------------- problem constants ----------------
#define NB   2
#define CCH  32
#define NCK  4
#define DIMH 96
#define SW   96                       // w stride (floats)
#define SD   (96*96)                  // d stride (floats)
#define PP   (96*96*96)               // 884736 voxels
#define NKP  (NB*NCK*PP)              // 7077888

typedef float v2f __attribute__((ext_vector_type(2)));
typedef float v8f __attribute__((ext_vector_type(8)));

__device__ __forceinline__ float sigf(float v) { return 1.0f / (1.0f + __expf(-v)); }

// ---------------- zero small accumulators ----------------
__global__ void k_zero(float* p, int n) {
    for (int i = threadIdx.x; i < n; i += blockDim.x) p[i] = 0.0f;
}

// ---------------- pooling: H axis (sigmoid fused) ----------------
__global__ void __launch_bounds__(256) k_pool_h(const float* __restrict__ x,
                                                float* __restrict__ hmax,
                                                float* __restrict__ hmin) {
    int i = blockIdx.x * blockDim.x + threadIdx.x;
    if (i >= NKP) return;
    int h = i % DIMH;
    float mx = -INFINITY, mn = INFINITY;
#pragma unroll
    for (int dh = -2; dh <= 2; dh++) {
        int hh = h + dh;
        if (hh >= 0 && hh < DIMH) {
            float v = sigf(x[i + dh]);
            mx = fmaxf(mx, v); mn = fminf(mn, v);
        }
    }
    hmax[i] = mx; hmin[i] = mn;
}

// ---------------- pooling: W axis ----------------
__global__ void __launch_bounds__(256) k_pool_w(const float* __restrict__ x,
                                                const float* __restrict__ hmax,
                                                float* __restrict__ hwmax,
                                                float* __restrict__ ero) {
    int i = blockIdx.x * blockDim.x + threadIdx.x;
    if (i >= NKP) return;
    int w = (i / SW) % DIMH;
    float mx = -INFINITY, mn = INFINITY;
#pragma unroll
    for (int dw = -2; dw <= 2; dw++) {
        int ww = w + dw;
        if (ww >= 0 && ww < DIMH) {
            int j = i + dw * SW;
            mx = fmaxf(mx, hmax[j]);
            mn = fminf(mn, sigf(x[j]));
        }
    }
    hwmax[i] = mx;
    ero[i] = fminf(ero[i], mn);
}

// ---------------- pooling: D axis (finalizes dil + ero) ----------------
__global__ void __launch_bounds__(256) k_pool_d(const float* __restrict__ x,
                                                const float* __restrict__ hwmax,
                                                float* __restrict__ dil,
                                                float* __restrict__ ero) {
    int i = blockIdx.x * blockDim.x + threadIdx.x;
    if (i >= NKP) return;
    int d = (i / SD) % DIMH;
    float mx = -INFINITY, mn = INFINITY;
#pragma unroll
    for (int dd = -2; dd <= 2; dd++) {
        int zz = d + dd;
        if (zz >= 0 && zz < DIMH) {
            int j = i + dd * SD;
            mx = fmaxf(mx, hwmax[j]);
            mn = fminf(mn, sigf(x[j]));
        }
    }
    dil[i] = mx;
    ero[i] = fminf(ero[i], mn);
}

// ---------------- cluster reduction: acc[n][c][{e0..3,d0..3,rowsum}] ----------------
__global__ void __launch_bounds__(256) k_cluster(const float* __restrict__ feat,
                                                 const float* __restrict__ ebuf,
                                                 const float* __restrict__ dbuf,
                                                 float* accG, float* sums) {
    int nc = blockIdx.y;
    int n = nc >> 5, c = nc & 31;
    const float* fr = feat + (size_t)(n * CCH + c) * PP;
    const float* er = ebuf + (size_t)n * NCK * PP;
    const float* dr = dbuf + (size_t)n * NCK * PP;
    float acc[9], sacc[8];
#pragma unroll
    for (int j = 0; j < 9; j++) acc[j] = 0.0f;
#pragma unroll
    for (int j = 0; j < 8; j++) sacc[j] = 0.0f;
    const bool doSums = (c == 0);
    for (int p = blockIdx.x * blockDim.x + threadIdx.x; p < PP; p += gridDim.x * blockDim.x) {
        float f = fr[p];
#pragma unroll
        for (int k = 0; k < 4; k++) {
            float ev = er[k * PP + p];
            float dv = dr[k * PP + p];
            acc[k]     += f * ev;
            acc[4 + k] += f * dv;
            if (doSums) { sacc[k] += ev; sacc[4 + k] += dv; }
        }
        acc[8] += f;
    }
#pragma unroll
    for (int j = 0; j < 9; j++)
#pragma unroll
        for (int ofs = 1; ofs < 32; ofs <<= 1) acc[j] += __shfl_xor(acc[j], ofs, 32);
#pragma unroll
    for (int j = 0; j < 8; j++)
#pragma unroll
        for (int ofs = 1; ofs < 32; ofs <<= 1) sacc[j] += __shfl_xor(sacc[j], ofs, 32);
    if ((threadIdx.x & 31) == 0) {
        float* ag = accG + (size_t)(n * CCH + c) * 16;
#pragma unroll
        for (int j = 0; j < 9; j++) atomicAdd(&ag[j], acc[j]);
        if (doSums)
#pragma unroll
            for (int j = 0; j < 8; j++) atomicAdd(&sums[n * 8 + j], sacc[j]);
    }
}

// ---------------- finalize clusters -> WMMA per-lane operand tables ----------------
__global__ void k_cfin(const float* __restrict__ accG, const float* __restrict__ sums,
                       const float* __restrict__ kscal,
                       float* bG1, float* aG2, float* cseed) {
    __shared__ float diffL[NB][NCK][CCH];
    __shared__ float baseL[NB][CCH];
    int tid = threadIdx.x;
    float k0 = kscal[0];
    if (tid < CCH) {
        int c = tid;
        for (int n = 0; n < NB; n++) {
            const float* ag = accG + (size_t)(n * CCH + c) * 16;
            float rs = ag[8];
            float bsum = 0.0f;
            for (int k = 0; k < 4; k++) {
                float es  = sums[n * 8 + k]     + 1e-5f;
                float dsm = sums[n * 8 + 4 + k] + 1e-5f;
                float bs  = (float)PP - sums[n * 8 + 4 + k] + 1e-5f;
                float ec = ag[k] / es;
                float dc = ag[4 + k] / dsm;
                float bc = (rs - ag[4 + k]) / bs;
                diffL[n][k][c] = ec + dc - bc;     // fore - back
                bsum += bc;
            }
            baseL[n][c] = bsum;
        }
    }
    __syncthreads();
    // GEMM1 B operand: B[s][j][lane] = diff[col=lo][c=4s+j+2hi] (cols>=4 zero)
    for (int idx = tid; idx < NB * 8 * 2 * 32; idx += blockDim.x) {
        int lane = idx & 31, j = (idx >> 5) & 1, s = (idx >> 6) & 7, n = (idx >> 9) & 1;
        int hi = lane >> 4, lo = lane & 15;
        int cc = 4 * s + j + 2 * hi;
        bG1[idx] = (lo < 4) ? diffL[n][lo][cc] : 0.0f;
    }
    // GEMM2 A operand: A[t][j][lane] = k0*diff[k=j+2hi][c=16t+lo]
    for (int idx = tid; idx < NB * 2 * 2 * 32; idx += blockDim.x) {
        int lane = idx & 31, j = (idx >> 5) & 1, t = (idx >> 6) & 1, n = (idx >> 7) & 1;
        int hi = lane >> 4, lo = lane & 15;
        aG2[idx] = k0 * diffL[n][j + 2 * hi][16 * t + lo];
    }
    // GEMM2 C seed: C[t][vi][lane] = k0*base[c=16t+vi+8hi]
    for (int idx = tid; idx < NB * 2 * 8 * 32; idx += blockDim.x) {
        int lane = idx & 31, vi = (idx >> 5) & 7, t = (idx >> 8) & 1, n = (idx >> 9) & 1;
        int hi = lane >> 4;
        cseed[idx] = k0 * baseL[n][16 * t + vi + 8 * hi];
    }
}

// ---------------- attention: WMMA f32 16x16x4, out_pre + BN stats ----------------
__global__ void __launch_bounds__(256) k_atten(const float* __restrict__ feat,
                                               const float* __restrict__ bG1,
                                               const float* __restrict__ aG2,
                                               const float* __restrict__ cseed,
                                               float* __restrict__ out,
                                               float* __restrict__ stats) {
    const int n    = blockIdx.y;
    const int lane = threadIdx.x & 31;
    const int hi   = lane >> 4, lo = lane & 15;
    const int wid  = blockIdx.x * (blockDim.x >> 5) + (threadIdx.x >> 5);
    const int nw   = gridDim.x * (blockDim.x >> 5);
    const float* featn = feat + (size_t)n * CCH * PP;
    float* outn = out + (size_t)n * CCH * PP;

    v2f Bt[8];
#pragma unroll
    for (int s = 0; s < 8; s++) {
        Bt[s].x = bG1[((n * 8 + s) * 2 + 0) * 32 + lane];
        Bt[s].y = bG1[((n * 8 + s) * 2 + 1) * 32 + lane];
    }
    v2f A2[2];
#pragma unroll
    for (int t = 0; t < 2; t++) {
        A2[t].x = aG2[((n * 2 + t) * 2 + 0) * 32 + lane];
        A2[t].y = aG2[((n * 2 + t) * 2 + 1) * 32 + lane];
    }
    v8f C0[2];
#pragma unroll
    for (int t = 0; t < 2; t++)
#pragma unroll
        for (int vi = 0; vi < 8; vi++)
            C0[t][vi] = cseed[((n * 2 + t) * 8 + vi) * 32 + lane];

    float accS[16], accQ[16];
#pragma unroll
    for (int i = 0; i < 16; i++) { accS[i] = 0.0f; accQ[i] = 0.0f; }

    const int ntile = PP / 16;
    for (int tile = wid; tile < ntile; tile += nw) {
        int p0 = tile * 16;
        // GEMM1: g[p,k] = sum_c feat[c,p]*diff[k,c]; M=p(16), K=c (8 steps of 4)
        v8f g = {};
#pragma unroll
        for (int s = 0; s < 8; s++) {
            const float* ap = featn + (size_t)(4 * s + 2 * hi) * PP + p0 + lo;
            v2f a; a.x = ap[0]; a.y = ap[PP];
            g = __builtin_amdgcn_wmma_f32_16x16x4_f32(false, a, false, Bt[s],
                                                      (short)0, g, false, false);
        }
        // 2-way softmax -> sigmoid
        float sg[8];
#pragma unroll
        for (int vi = 0; vi < 8; vi++) sg[vi] = 1.0f / (1.0f + __expf(-g[vi]));
        // transpose s from D-layout (p across vgpr+half, k across lanes) to B-layout
        int src0 = 2 * hi + ((lo & 8) << 1);   // source lane = k + 16*(p_local>=8), k=2hi (j=0)
        int src1 = src0 + 1;                   // k=1+2hi (j=1)
        float b0 = 0.0f, b1 = 0.0f;
#pragma unroll
        for (int vi = 0; vi < 8; vi++) {
            float t0 = __shfl(sg[vi], src0, 32);
            float t1 = __shfl(sg[vi], src1, 32);
            bool sel = ((lo & 7) == vi);
            b0 = sel ? t0 : b0;
            b1 = sel ? t1 : b1;
        }
        v2f bs; bs.x = b0; bs.y = b1;
        // GEMM2: atten[c,p] = k0*base[c] + sum_k k0*diff[k,c]*s[k,p]; single K=4 WMMA per c-tile
#pragma unroll
        for (int t = 0; t < 2; t++) {
            v8f dt = __builtin_amdgcn_wmma_f32_16x16x4_f32(false, A2[t], false, bs,
                                                           (short)0, C0[t], false, false);
#pragma unroll
            for (int vi = 0; vi < 8; vi++) {
                int c = 16 * t + vi + 8 * hi;
                size_t ad = (size_t)c * PP + p0 + lo;
                float f = featn[ad];
                float o = 5.0f * f + dt[vi];     // (1+NC)*feat + k*atten
                outn[ad] = o;
                accS[t * 8 + vi] += o;
                accQ[t * 8 + vi] += o * o;
            }
        }
    }
    // reduce stats within each 16-lane half (channel fixed per (t,vi,hi))
#pragma unroll
    for (int i = 0; i < 16; i++) {
#pragma unroll
        for (int ofs = 1; ofs < 16; ofs <<= 1) {
            accS[i] += __shfl_xor(accS[i], ofs, 32);
            accQ[i] += __shfl_xor(accQ[i], ofs, 32);
        }
    }
    if (lo == 0) {
#pragma unroll
        for (int t = 0; t < 2; t++)
#pragma unroll
            for (int vi = 0; vi < 8; vi++) {
                int c = 16 * t + vi + 8 * hi;
                atomicAdd(&stats[c],      accS[t * 8 + vi]);
                atomicAdd(&stats[32 + c], accQ[t * 8 + vi]);
            }
    }
}

// ---------------- BN finalize + apply ----------------
__global__ void k_bnfin(const float* __restrict__ stats, const float* __restrict__ bnw,
                        const float* __restrict__ bnb, float* bnab) {
    int c = threadIdx.x;
    if (c >= CCH) return;
    float cnt = (float)NB * (float)PP;
    float mean = stats[c] / cnt;
    float var  = stats[32 + c] / cnt - mean * mean;
    float sc   = bnw[c] * rsqrtf(var + 1e-5f);
    bnab[c]      = sc;
    bnab[32 + c] = bnb[c] - mean * sc;
}

__global__ void __launch_bounds__(256) k_bnapply(float* __restrict__ out,
                                                 const float* __restrict__ bnab) {
    int i = blockIdx.x * blockDim.x + threadIdx.x;
    const int n4 = NB * CCH * (PP / 4);
    if (i >= n4) return;
    int c = (i / (PP / 4)) % CCH;
    float sc = bnab[c], sh = bnab[32 + c];
    float4 v = ((float4*)out)[i];
    v.x = v.x * sc + sh; v.y = v.y * sc + sh;
    v.z = v.z * sc + sh; v.w = v.w * sc + sh;
    ((float4*)out)[i] = v;
}

// ---------------- host launcher ----------------
extern "C" void kernel_launch(void* const* d_in, const int* in_sizes, int n_in,
                              void* d_out, int out_size, void* d_ws, size_t ws_size,
                              hipStream_t stream) {
    const float* feat = (const float*)d_in[0];
    const float* x    = (const float*)d_in[1];
    const float* kin  = (const float*)d_in[2];
    const float* bnw  = (const float*)d_in[3];
    const float* bnb  = (const float*)d_in[4];
    float* out = (float*)d_out;
    float* ws  = (float*)d_ws;

    float* B1    = ws;                          // hmax -> dil
    float* B2    = ws + (size_t)NKP;            // hmin -> ero
    float* B3    = ws + 2 * (size_t)NKP;        // hwmax
    float* accG  = ws + 3 * (size_t)NKP;        // [2][32][16]
    float* sums  = accG + 1024;                 // [2][8]
    float* stats = sums + 16;                   // [64]
    float* bG1   = stats + 64;                  // [2][8][2][32]
    float* aG2   = bG1 + 1024;                  // [2][2][2][32]
    float* cseed = aG2 + 256;                   // [2][2][8][32]
    float* bnab  = cseed + 1024;                // [64]

    k_zero<<<1, 256, 0, stream>>>(accG, 1024 + 16 + 64);

    k_pool_h<<<NKP / 256, 256, 0, stream>>>(x, B1, B2);
    k_pool_w<<<NKP / 256, 256, 0, stream>>>(x, B1, B3, B2);
    k_pool_d<<<NKP / 256, 256, 0, stream>>>(x, B3, B1, B2);

    dim3 g4(54, NB * CCH);
    k_cluster<<<g4, 256, 0, stream>>>(feat, B2, B1, accG, sums);

    k_cfin<<<1, 256, 0, stream>>>(accG, sums, kin, bG1, aG2, cseed);

    dim3 g6(288, NB);
    k_atten<<<g6, 256, 0, stream>>>(feat, bG1, aG2, cseed, out, stats);

    k_bnfin<<<1, 32, 0, stream>>>(stats, bnw, bnb, bnab);
    k_bnapply<<<NB * CCH * (PP / 4) / 256, 256, 0, stream>>>(out, bnab);
}